// WeightedGCN_68582037782885
// MI455X (gfx1250) — compile-verified
//
#include <hip/hip_runtime.h>
#include <hip/hip_bf16.h>
#include <math.h>

#define N_NODES   100000
#define N_EDGES   6400000
#define N_FEAT    128
#define HIDDEN    16
#define N_CLASSES 16
#define N_TILES   (N_NODES / 16)   // 6250, exact

typedef __attribute__((ext_vector_type(2))) float v2f;
typedef __attribute__((ext_vector_type(8))) float v8f;

// ---------------------------------------------------------------- degrees ---
__global__ void deg_init_kernel(float* __restrict__ deg) {
    int i = blockIdx.x * blockDim.x + threadIdx.x;
    if (i < N_NODES) deg[i] = 1.0f;   // self-loop weight
}

__global__ void deg_edge_kernel(const long long* __restrict__ ei,
                                const float* __restrict__ w,
                                float* __restrict__ deg) {
    int e = blockIdx.x * blockDim.x + threadIdx.x;
    if (e < N_EDGES) {
        int d = (int)ei[N_EDGES + e];          // row 1 of [2,E] = dst
        atomicAdd(&deg[d], w[e]);
    }
}

__global__ void dinv_kernel(float* __restrict__ deg) {
    int i = blockIdx.x * blockDim.x + threadIdx.x;
    if (i < N_NODES) {
        float dg = deg[i];
        deg[i] = (dg > 0.0f) ? rsqrtf(dg) : 0.0f;   // in place: deg -> dinv
    }
}

// ------------------------------------------------- layer 1: x @ W1 (WMMA) ---
// One wave computes a 16x16 output tile (16 node rows x HIDDEN).
// Epilogue: hs = h*dinv (src-side factor), out1 = b1 + hs*dinv (self loop + bias).
__global__ void gemm1_wmma_kernel(const float* __restrict__ x,
                                  const float* __restrict__ W1,
                                  const float* __restrict__ b1,
                                  const float* __restrict__ dinv,
                                  float* __restrict__ hs,
                                  float* __restrict__ out1) {
    const int lane = threadIdx.x & 31;
    const int wave = threadIdx.x >> 5;
    const int tile = blockIdx.x * (blockDim.x >> 5) + wave;
    if (tile >= N_TILES) return;            // wave-uniform: EXEC stays all-1s
    const int half = lane >> 4;             // 0: K pair {k,k+1}; 1: {k+2,k+3}
    const int l16  = lane & 15;

    const size_t arow = (size_t)(tile * 16 + l16) * N_FEAT;
    v8f c = {};
#pragma unroll
    for (int k = 0; k < N_FEAT; k += 4) {
        const int kk = k + 2 * half;
        v2f a = *(const v2f*)(x + arow + kk);               // A[m][kk], A[m][kk+1]
        v2f b;
        b.x = W1[(size_t)kk * HIDDEN + l16];                // B[kk][n]
        b.y = W1[(size_t)(kk + 1) * HIDDEN + l16];          // B[kk+1][n]
        c = __builtin_amdgcn_wmma_f32_16x16x4_f32(
                false, a, false, b, (short)0, c, false, false);
    }

    const float bn = b1[l16];
#pragma unroll
    for (int v = 0; v < 8; ++v) {
        const int   r  = tile * 16 + v + 8 * half;          // C/D row mapping
        const float dv = dinv[r];
        const float hv = c[v] * dv;                         // h * dinv[src]
        hs[(size_t)r * HIDDEN + l16]   = hv;
        out1[(size_t)r * HIDDEN + l16] = bn + hv * dv;      // bias + self-loop
    }
}

// ---------------------------------------- edge scatter-add (16 lanes/edge) ---
__global__ void edge_agg_kernel(const long long* __restrict__ ei,
                                const float* __restrict__ w,
                                const float* __restrict__ dinv,
                                const float* __restrict__ hs,
                                float* __restrict__ out) {
    long long t = (long long)blockIdx.x * blockDim.x + threadIdx.x;
    long long e = t >> 4;
    int       f = (int)(t & 15);
    if (e >= N_EDGES) return;
    int s = (int)ei[e];
    int d = (int)ei[N_EDGES + e];
    float sc = w[e] * dinv[d];                               // dinv[src] folded in hs
    atomicAdd(out + (size_t)d * 16 + f, hs[(size_t)s * 16 + f] * sc);
}

// -------------------------------------- layer 2: relu(out1) @ W2 (WMMA) ------
__global__ void gemm2_wmma_kernel(const float* __restrict__ out1,
                                  const float* __restrict__ W2,
                                  const float* __restrict__ b2,
                                  const float* __restrict__ dinv,
                                  float* __restrict__ hs,
                                  float* __restrict__ out2) {
    const int lane = threadIdx.x & 31;
    const int wave = threadIdx.x >> 5;
    const int tile = blockIdx.x * (blockDim.x >> 5) + wave;
    if (tile >= N_TILES) return;
    const int half = lane >> 4;
    const int l16  = lane & 15;

    const size_t arow = (size_t)(tile * 16 + l16) * HIDDEN;
    v8f c = {};
#pragma unroll
    for (int k = 0; k < HIDDEN; k += 4) {
        const int kk = k + 2 * half;
        v2f a = *(const v2f*)(out1 + arow + kk);
        a.x = fmaxf(a.x, 0.0f);                              // fused ReLU
        a.y = fmaxf(a.y, 0.0f);
        v2f b;
        b.x = W2[(size_t)kk * N_CLASSES + l16];
        b.y = W2[(size_t)(kk + 1) * N_CLASSES + l16];
        c = __builtin_amdgcn_wmma_f32_16x16x4_f32(
                false, a, false, b, (short)0, c, false, false);
    }

    const float bn = b2[l16];
#pragma unroll
    for (int v = 0; v < 8; ++v) {
        const int   r  = tile * 16 + v + 8 * half;
        const float dv = dinv[r];
        const float hv = c[v] * dv;
        hs[(size_t)r * N_CLASSES + l16]   = hv;
        out2[(size_t)r * N_CLASSES + l16] = bn + hv * dv;
    }
}

// ------------------------------------------------ log_softmax (16 classes) ---
__global__ void log_softmax_kernel(float* __restrict__ out) {
    int t = blockIdx.x * blockDim.x + threadIdx.x;
    int node = t >> 4;
    int f    = t & 15;
    if (node >= N_NODES) return;
    float v = out[(size_t)node * 16 + f];
    float m = v;
#pragma unroll
    for (int off = 8; off; off >>= 1) m = fmaxf(m, __shfl_xor(m, off, 16));
    float s = expf(v - m);
#pragma unroll
    for (int off = 8; off; off >>= 1) s += __shfl_xor(s, off, 16);
    out[(size_t)node * 16 + f] = v - m - logf(s);
}

// ----------------------------------------------------------------- launch ---
extern "C" void kernel_launch(void* const* d_in, const int* in_sizes, int n_in,
                              void* d_out, int out_size, void* d_ws, size_t ws_size,
                              hipStream_t stream) {
    (void)in_sizes; (void)n_in; (void)out_size; (void)ws_size;

    const float*     x  = (const float*)d_in[0];
    const long long* ei = (const long long*)d_in[1];   // [2, E] int64
    const float*     ew = (const float*)d_in[2];
    const float*     W1 = (const float*)d_in[3];
    const float*     b1 = (const float*)d_in[4];
    const float*     W2 = (const float*)d_in[5];
    const float*     b2 = (const float*)d_in[6];
    float*           out = (float*)d_out;

    float* ws   = (float*)d_ws;
    float* dinv = ws;                                  // N floats (deg -> dinv)
    float* hs   = ws + N_NODES;                        // 16N floats (both layers)
    float* out1 = ws + N_NODES + 16 * (size_t)N_NODES; // 16N floats

    const int nodeBlocks = (N_NODES + 255) / 256;
    const int edgeBlocks = (N_EDGES + 255) / 256;
    const long long aggThreads = (long long)N_EDGES * 16;
    const int aggBlocks  = (int)((aggThreads + 255) / 256);
    const int gemmBlocks = (N_TILES + 7) / 8;          // 8 waves / 256-thread block
    const int lsmBlocks  = (N_NODES * 16 + 255) / 256;

    // degrees -> dinv
    deg_init_kernel<<<nodeBlocks, 256, 0, stream>>>(dinv);
    deg_edge_kernel<<<edgeBlocks, 256, 0, stream>>>(ei, ew, dinv);
    dinv_kernel<<<nodeBlocks, 256, 0, stream>>>(dinv);

    // layer 1
    gemm1_wmma_kernel<<<gemmBlocks, 256, 0, stream>>>(x, W1, b1, dinv, hs, out1);
    edge_agg_kernel<<<aggBlocks, 256, 0, stream>>>(ei, ew, dinv, hs, out1);

    // layer 2 (ReLU fused into A-fragment load)
    gemm2_wmma_kernel<<<gemmBlocks, 256, 0, stream>>>(out1, W2, b2, dinv, hs, out);
    edge_agg_kernel<<<aggBlocks, 256, 0, stream>>>(ei, ew, dinv, hs, out);

    // log_softmax in place
    log_softmax_kernel<<<lsmBlocks, 256, 0, stream>>>(out);
}